// SignatureTransform_46394236731641
// MI455X (gfx1250) — compile-verified
//
#include <hip/hip_runtime.h>

// ---------------------------------------------------------------------------
// Depth-4 path signature on gfx1250 (MI455X).
//
// Strategy:
//  * Chen's identity is associative -> split each path into NSEG segments,
//    compute segment signatures in parallel, then fold with the tensor-
//    algebra product. Removes the L=2048 sequential dependence.
//  * Inside a segment, level-4 accumulation is exactly a (512 x T)*(T x 8)
//    fp32 GEMM over per-step "G" columns:
//       C4 = sum_t G_t (x) z_t,
//       G = A3 + (A2(x)z)/2 + (A1(x)z(x)z)/6 + z(x)z(x)z/24
//    -> mapped onto V_WMMA_F32_16X16X4_F32 (f32 in/out: preserves the fp32
//    reference precision, unlike f16/fp8 WMMA).
//  * Levels 1-3 are the cheap sequential prefix (LDS + barriers) that feeds G.
// ---------------------------------------------------------------------------

typedef __attribute__((ext_vector_type(2))) float v2f;
typedef __attribute__((ext_vector_type(8))) float v8f;

#define BATCH   64
#define LPTS    2048
#define DCH     8
#define NSEG    32
#define TSEG    64          // increments per segment (last segment: 63)
#define CHUNK   16          // K-chunk staged in LDS
#define NCHUNK  (TSEG / CHUNK)
#define SIG_LEN 4680        // 8 + 64 + 512 + 4096
#define OFF2    8
#define OFF3    72
#define OFF4    584
#define GSTRIDE (CHUNK + 1) // 17: pad to avoid LDS bank conflicts

// ---------------------------------------------------------------------------
// Kernel 1: one block (256 thr = 8 waves) per (batch, segment).
// ---------------------------------------------------------------------------
__global__ __launch_bounds__(256) void seg_sig_kernel(
    const float* __restrict__ path, float* __restrict__ ws) {
  __shared__ float zbuf[TSEG * DCH];        // increments, zero-padded
  __shared__ float Gbuf[512 * GSTRIDE];     // G columns for one K-chunk
  __shared__ float A1[8], A2[64], A3[512];  // prefix state (levels 1..3)
  __shared__ float Hbuf[64], Pbuf[64];

  const int tid  = threadIdx.x;
  const int blk  = blockIdx.x;
  const int b    = blk / NSEG;
  const int s    = blk % NSEG;
  const int t0   = s * TSEG;
  const int rem  = (LPTS - 1) - t0;
  const int Tloc = rem < TSEG ? rem : TSEG;

  const int wave = tid >> 5;
  const int lane = tid & 31;
  const int half = lane >> 4;   // WMMA lane-half
  const int idx  = lane & 15;   // row (A) / col (B,D) within tile

  // Load increments z_t = path[t+1]-path[t]; pad with 0 (zero increment is
  // an exact no-op in every formula below).
  const float* pb = path + (size_t)b * LPTS * DCH;
  for (int x = tid; x < TSEG * DCH; x += 256) {
    int t = x >> 3, d = x & 7;
    float z = 0.f;
    if (t < Tloc) z = pb[(t0 + t + 1) * DCH + d] - pb[(t0 + t) * DCH + d];
    zbuf[x] = z;
  }
  for (int x = tid; x < 512; x += 256) A3[x] = 0.f;
  if (tid < 64) A2[tid] = 0.f;
  if (tid < 8)  A1[tid] = 0.f;
  __syncthreads();

  // Level-4 WMMA accumulators: 32 row-tiles / 8 waves = 4 tiles per wave.
  v8f acc[4];
#pragma unroll
  for (int q = 0; q < 4; ++q)
#pragma unroll
    for (int e = 0; e < 8; ++e) acc[q][e] = 0.f;

  for (int c = 0; c < NCHUNK; ++c) {
    // ---- sequential prefix for CHUNK steps; emit G columns ----
    for (int u = 0; u < CHUNK; ++u) {
      const float* z = &zbuf[(c * CHUNK + u) * DCH];
      // Stage 1: everything reading PRE-update state.
      if (tid < 64) {
        int i = tid >> 3, j = tid & 7;
        float zi = z[i], zj = z[j];
        Hbuf[tid] = A2[tid] + 0.5f * A1[i] * zj + (1.f / 6.f) * zi * zj;
        Pbuf[tid] = (A1[i] + 0.5f * zi) * zj;
      }
      for (int x = tid; x < 512; x += 256) {
        int i = x >> 6, j = (x >> 3) & 7, k = x & 7;
        float zj = z[j], zk = z[k];
        float g = A3[x] + 0.5f * A2[x >> 3] * zk +
                  (1.f / 6.f) * A1[i] * zj * zk +
                  (1.f / 24.f) * z[i] * zj * zk;
        Gbuf[x * GSTRIDE + u] = g;
      }
      __syncthreads();
      // Stage 2: state update (independent writes).
      for (int x = tid; x < 512; x += 256) A3[x] += Hbuf[x >> 3] * z[x & 7];
      if (tid < 64) A2[tid] += Pbuf[tid];
      if (tid >= 64 && tid < 72) A1[tid - 64] += z[tid - 64];
      __syncthreads();
    }

    // ---- WMMA: C4 += Gmat(512 x 16) * Zmat(16 x 8, padded to 16 cols) ----
    // A frag: lane(half,idx) vgpr j holds A[idx][2*half+j]
    // B frag: lane(half,idx) vgpr j holds B[2*half+j][idx]
#pragma unroll
    for (int q = 0; q < 4; ++q) {
      const int tile = wave + 8 * q;
      const int row  = tile * 16 + idx;
#pragma unroll
      for (int c4 = 0; c4 < 4; ++c4) {
        const int kk = c4 * 4 + 2 * half;      // chunk-local K index
        v2f a, bf;
        a[0] = Gbuf[row * GSTRIDE + kk];
        a[1] = Gbuf[row * GSTRIDE + kk + 1];
        const int tg = c * CHUNK + kk;         // global step index
        bf[0] = (idx < 8) ? zbuf[tg * DCH + idx] : 0.f;
        bf[1] = (idx < 8) ? zbuf[(tg + 1) * DCH + idx] : 0.f;
        acc[q] = __builtin_amdgcn_wmma_f32_16x16x4_f32(
            false, a, false, bf, (short)0, acc[q], false, false);
      }
    }
    __syncthreads();  // before next chunk overwrites Gbuf
  }

  // ---- write segment signature: [A1 | A2 | A3 | C4] ----
  float* sig = ws + (size_t)blk * SIG_LEN;
  if (tid < 8)  sig[tid] = A1[tid];
  if (tid < 64) sig[OFF2 + tid] = A2[tid];
  for (int x = tid; x < 512; x += 256) sig[OFF3 + x] = A3[x];
  if (idx < 8) {
#pragma unroll
    for (int q = 0; q < 4; ++q) {
      const int tile = wave + 8 * q;
#pragma unroll
      for (int v = 0; v < 8; ++v) {
        const int row = tile * 16 + v + 8 * half;  // D vgpr v -> M = v + 8*half
        sig[OFF4 + row * 8 + idx] = acc[q][v];
      }
    }
  }
}

// ---------------------------------------------------------------------------
// Kernel 2: one block per batch; left-to-right tensor-algebra fold:
//   C_k = sum_{j=0..k} R_{k-j} (x) B_j   (R_0 = B_0 = 1)
// ---------------------------------------------------------------------------
__global__ __launch_bounds__(256) void combine_kernel(
    const float* __restrict__ ws, float* __restrict__ out) {
  __shared__ float R[SIG_LEN];
  __shared__ float Bb[SIG_LEN];
  const int tid = threadIdx.x;
  const int b   = blockIdx.x;
  const float* base = ws + (size_t)b * NSEG * SIG_LEN;

  for (int x = tid; x < SIG_LEN; x += 256) R[x] = base[x];
  __syncthreads();

  for (int s = 1; s < NSEG; ++s) {
    const float* seg = base + (size_t)s * SIG_LEN;
    for (int x = tid; x < SIG_LEN; x += 256) Bb[x] = seg[x];
    __syncthreads();
    float *R1 = R, *R2 = R + OFF2, *R3 = R + OFF3, *R4 = R + OFF4;
    float *B1 = Bb, *B2 = Bb + OFF2, *B3 = Bb + OFF3, *B4 = Bb + OFF4;
    // level 4 (reads R1..R3 pre-update)
    for (int x = tid; x < 4096; x += 256) {
      R4[x] += B4[x] + R3[x >> 3] * B1[x & 7] + R2[x >> 6] * B2[x & 63] +
               R1[x >> 9] * B3[x & 511];
    }
    __syncthreads();
    // level 3
    for (int x = tid; x < 512; x += 256) {
      R3[x] += B3[x] + R2[x >> 3] * B1[x & 7] + R1[x >> 6] * B2[x & 63];
    }
    __syncthreads();
    if (tid < 64) R2[tid] += B2[tid] + R1[tid >> 3] * B1[tid & 7];
    __syncthreads();
    if (tid < 8) R1[tid] += B1[tid];
    __syncthreads();
  }

  float* o = out + (size_t)b * SIG_LEN;
  for (int x = tid; x < SIG_LEN; x += 256) o[x] = R[x];
}

// ---------------------------------------------------------------------------
extern "C" void kernel_launch(void* const* d_in, const int* in_sizes, int n_in,
                              void* d_out, int out_size, void* d_ws,
                              size_t ws_size, hipStream_t stream) {
  const float* path = (const float*)d_in[0];
  float* out = (float*)d_out;
  float* ws  = (float*)d_ws;  // needs 64*32*4680*4 B = ~38.3 MB

  seg_sig_kernel<<<BATCH * NSEG, 256, 0, stream>>>(path, ws);
  combine_kernel<<<BATCH, 256, 0, stream>>>(ws, out);
}